// TopLoss_7060926235055
// MI455X (gfx1250) — compile-verified
//
#include <hip/hip_runtime.h>
#include <stdint.h>

// -------- CDNA5 WMMA vector types (wave32) --------
typedef __attribute__((ext_vector_type(2))) float v2f;   // A/B frag of 16x16x4 f32 (64 vals / 32 lanes)
typedef __attribute__((ext_vector_type(8))) float v8f;   // 16x16 f32 C/D frag

#define NCAP 10240   // max simplices held in LDS piv[] (n = 9843 for the fixed 8x8x8 grid)
#define WCAP 384     // max packed row words (W32 = ceil(9843/32) = 308)

// float -> uint32 key that is strictly DECREASING in the float value
// (so ascending uint sort == descending f == ascending (-f), matching lexsort)
__device__ __forceinline__ uint32_t f32_desc_key(float x) {
  uint32_t u = __float_as_uint(x);
  uint32_t asc = (u & 0x80000000u) ? ~u : (u | 0x80000000u);  // ascending map
  return ~asc;                                                // descending map
}

// ---- K1: filtration value f[j] = min over 4 verts; build composite sort keys ----
__global__ void k_filtration(const float* __restrict__ beta, const int* __restrict__ sv,
                             const int* __restrict__ sdim, int n, int npad,
                             float* __restrict__ f, uint64_t* __restrict__ keys) {
  int i = blockIdx.x * blockDim.x + threadIdx.x;
  if (i >= npad) return;
  if (i >= n) { keys[i] = ~0ull; return; }   // padding sorts to the end
  float a = beta[sv[4 * i + 0]];
  float b = beta[sv[4 * i + 1]];
  float c = beta[sv[4 * i + 2]];
  float d = beta[sv[4 * i + 3]];
  float mn = fminf(fminf(a, b), fminf(c, d));
  f[i] = mn;
  // key bits: [63:32] desc(f)  [31:30] dim  [29:0] original index (stable tiebreak)
  keys[i] = ((uint64_t)f32_desc_key(mn) << 32) |
            ((uint64_t)(sdim[i] & 3) << 30) |
            (uint32_t)i;
}

// ---- K2: one bitonic compare-exchange stage (ascending overall) ----
__global__ void k_bitonic(uint64_t* __restrict__ keys, int jj, int k, int npad) {
  int i = blockIdx.x * blockDim.x + threadIdx.x;
  if (i >= npad) return;
  int ixj = i ^ jj;
  if (ixj > i) {
    uint64_t a = keys[i], b = keys[ixj];
    if (((i & k) == 0) ? (a > b) : (a < b)) { keys[i] = b; keys[ixj] = a; }
  }
}

// ---- K3: unpack sorted keys -> inv permutation, sorted filtration values & dims ----
__global__ void k_unpack(const uint64_t* __restrict__ keys, const float* __restrict__ f,
                         const int* __restrict__ sdim, int n,
                         int* __restrict__ inv, float* __restrict__ fs, int* __restrict__ dsort) {
  int i = blockIdx.x * blockDim.x + threadIdx.x;
  if (i >= n) return;
  int p = (int)(keys[i] & 0x3FFFFFFFu);   // original simplex id
  inv[p] = i;
  fs[i] = f[p];
  dsort[i] = sdim[p];
}

// ---- K4a: zero the packed boundary matrix ----
__global__ void k_zero(uint32_t* __restrict__ B, size_t nwords) {
  size_t i = (size_t)blockIdx.x * blockDim.x + threadIdx.x;
  if (i < nwords) B[i] = 0u;
}

// ---- K4b: scatter boundary incidences: B[inv[cof]][inv[face]] = 1 (bit-packed) ----
__global__ void k_scatter(const int* __restrict__ inv, const int* __restrict__ bc,
                          const int* __restrict__ bf, int m, int W32,
                          uint32_t* __restrict__ B) {
  int i = blockIdx.x * blockDim.x + threadIdx.x;
  if (i >= m) return;
  int r = inv[bc[i]];
  int c = inv[bf[i]];
  atomicOr(&B[(size_t)r * W32 + (c >> 5)], 1u << (c & 31));
}

// low 32 bits of a generic pointer to LDS == byte offset within the wave's LDS
__device__ __forceinline__ uint32_t lds_addr32(const void* p) {
  return (uint32_t)(uintptr_t)p;
}

// ---- K5: sequential persistence reduction, single workgroup ----
// Double-buffered rows in LDS: row j+1 streams in via the CDNA5 async
// global->LDS engine (ASYNCcnt) while row j is being reduced.
__global__ void __launch_bounds__(256) k_reduce(uint32_t* __restrict__ B, int n, int W32,
                                                int* __restrict__ piv_out) {
  __shared__ int piv[NCAP];
  __shared__ __attribute__((aligned(16))) uint32_t rowbuf[2][WCAP];
  __shared__ int red[256];
  if (n > NCAP || W32 > WCAP) return;
  const int t = threadIdx.x;
  const int nchunk = W32 >> 2;                 // 16-byte async chunks per row
  const bool async_ok = ((W32 & 3) == 0);      // row base stays 16B-aligned (true: 308%4==0)

  for (int i = t; i < n; i += 256) piv[i] = -1;

  // preload row 0 into rowbuf[0]
  if (async_ok) {
    if (t < nchunk) {
      uint32_t lds = lds_addr32(&rowbuf[0][t << 2]);
      const uint32_t* g = B + (t << 2);
      asm volatile("global_load_async_to_lds_b128 %0, %1, off"
                   :: "v"(lds), "v"(g) : "memory");
    }
  } else {
    for (int w = t; w < W32; w += 256) rowbuf[0][w] = B[w];
  }

  for (int j = 0; j < n; ++j) {
    uint32_t* wr = rowbuf[j & 1];
    asm volatile("s_wait_asynccnt 0x0" ::: "memory");  // row j resident in LDS
    __syncthreads();

    // kick off async stream of row j+1 into the other buffer (race-free:
    // B rows are only rewritten at their own iteration's store-back)
    if (j + 1 < n) {
      uint32_t* nb = rowbuf[(j + 1) & 1];
      const uint32_t* Bn = B + (size_t)(j + 1) * W32;
      if (async_ok) {
        if (t < nchunk) {
          uint32_t lds = lds_addr32(&nb[t << 2]);
          const uint32_t* g = Bn + (t << 2);
          asm volatile("global_load_async_to_lds_b128 %0, %1, off"
                       :: "v"(lds), "v"(g) : "memory");
        }
      } else {
        for (int w = t; w < W32; w += 256) nb[w] = Bn[w];
      }
    }

    for (;;) {
      // low(row): max set bit index (parallel max-reduce), -1 if row empty
      int local = -1;
      for (int w = t; w < W32; w += 256) {
        uint32_t u = wr[w];
        if (u) { int c = (w << 5) + (31 - __clz(u)); if (c > local) local = c; }
      }
      red[t] = local;
      __syncthreads();
      for (int s = 128; s > 0; s >>= 1) {
        if (t < s) { int o = red[t + s]; if (o > red[t]) red[t] = o; }
        __syncthreads();
      }
      int low = red[0];
      __syncthreads();              // protect red[] before next pass rewrites it
      if (low < 0) break;           // uniform
      int p = piv[low];             // uniform shared read
      __syncthreads();              // all reads done before the (possible) write below
      if (p < 0) {
        if (t == 0) piv[low] = j;
        break;                      // uniform
      }
      const uint32_t* Bp = B + (size_t)p * W32;
      for (int w = t; w < W32; w += 256) {
        __builtin_prefetch(&Bp[w], 0, 1);   // global_prefetch path
        wr[w] ^= Bp[w];
      }
      __syncthreads();
    }
    __syncthreads();                 // piv write visible, row stable
    uint32_t* Bj = B + (size_t)j * W32;
    for (int w = t; w < W32; w += 256) Bj[w] = wr[w];  // reduced row may become a pivot later
    __syncthreads();
  }
  for (int i = t; i < n; i += 256) piv_out[i] = piv[i];
}

// ---- K6: barcode-length sums via fp32 WMMA reduction engine ----
// wave 0 sums dim-0 bar lengths, wave 1 sums dim-1 bar lengths + tracks max.
// A (16x4) holds 64 masked lengths per tile, B = all-ones => D accumulates exact fp32 sums.
__global__ void __launch_bounds__(64) k_loss(const float* __restrict__ fs,
                                             const int* __restrict__ dsort,
                                             const int* __restrict__ piv, int n,
                                             float* __restrict__ out) {
  __shared__ float part[64];
  __shared__ float mx[32];
  const int t = threadIdx.x;
  const int wave = t >> 5;     // 0 or 1
  const int lane = t & 31;
  const int dim = wave;        // wave w handles homology dimension w

  v8f acc = {};
  v2f bmat;
  bmat.x = 1.0f;
  bmat.y = 1.0f;

  const int ntiles = (n + 63) >> 6;
  float localmax = 0.0f;
  for (int tile = 0; tile < ntiles; ++tile) {
    int i0 = tile * 64 + lane * 2;
    int i1 = i0 + 1;
    float e0 = 0.0f, e1 = 0.0f;
    if (i0 < n) {
      int p = piv[i0];
      float len = (p >= 0) ? (fs[i0] - fs[p]) : 0.0f;
      e0 = (dsort[i0] == dim) ? len : 0.0f;
    }
    if (i1 < n) {
      int p = piv[i1];
      float len = (p >= 0) ? (fs[i1] - fs[p]) : 0.0f;
      e1 = (dsort[i1] == dim) ? len : 0.0f;
    }
    localmax = fmaxf(localmax, fmaxf(e0, e1));
    v2f amat;
    amat.x = e0;
    amat.y = e1;
    // D = A x ones + C  : exact fp32 accumulation in the matrix pipe
    acc = __builtin_amdgcn_wmma_f32_16x16x4_f32(false, amat, false, bmat,
                                                (short)0, acc, false, false);
  }
  // column 0 of D lives in lane 0 (M=0..7) and lane 16 (M=8..15)
  float partial = acc[0] + acc[1] + acc[2] + acc[3] + acc[4] + acc[5] + acc[6] + acc[7];
  part[t] = partial;
  if (wave == 1) mx[lane] = localmax;
  __syncthreads();
  if (t == 0) {
    float sum0 = part[0] + part[16];     // loss0: sum of dim-0 bar lengths
    float sum1 = part[32] + part[48];    // sum of dim-1 bar lengths
    float m = 0.0f;
    for (int l = 0; l < 32; ++l) m = fmaxf(m, mx[l]);
    out[0] = sum0 + sum1 - m;            // loss0 + (sum1 - max1)
  }
}

extern "C" void kernel_launch(void* const* d_in, const int* in_sizes, int n_in,
                              void* d_out, int out_size, void* d_ws, size_t ws_size,
                              hipStream_t stream) {
  const float* beta = (const float*)d_in[0];
  const int* sv   = (const int*)d_in[1];
  const int* sdim = (const int*)d_in[2];
  const int* bc   = (const int*)d_in[3];
  const int* bf   = (const int*)d_in[4];
  const int n = in_sizes[2];     // number of simplices (9843)
  const int m = in_sizes[3];     // number of boundary incidences (27188)
  const int W32 = (n + 31) >> 5; // packed words per row (308)
  int npad = 1;
  while (npad < n) npad <<= 1;   // 16384

  // workspace layout
  char* base = (char*)d_ws;
  size_t off = 0;
  auto take = [&](size_t bytes) -> char* {
    char* p = base + off;
    off = (off + bytes + 15) & ~(size_t)15;
    return p;
  };
  uint64_t* keys = (uint64_t*)take((size_t)npad * 8);
  float*    f    = (float*)take((size_t)n * 4);
  int*      inv  = (int*)take((size_t)n * 4);
  float*    fs   = (float*)take((size_t)n * 4);
  int*      dsrt = (int*)take((size_t)n * 4);
  int*      piv  = (int*)take((size_t)n * 4);
  uint32_t* B    = (uint32_t*)take((size_t)n * W32 * 4);   // ~12.1 MB, L2-resident
  (void)ws_size; (void)n_in; (void)out_size;

  // 1) filtration values + sort keys
  k_filtration<<<(npad + 255) / 256, 256, 0, stream>>>(beta, sv, sdim, n, npad, f, keys);

  // 2) bitonic sort (ascending composite key == reference lexsort order)
  for (int k = 2; k <= npad; k <<= 1)
    for (int jj = k >> 1; jj > 0; jj >>= 1)
      k_bitonic<<<(npad + 255) / 256, 256, 0, stream>>>(keys, jj, k, npad);

  // 3) inverse permutation + sorted f/dim
  k_unpack<<<(n + 255) / 256, 256, 0, stream>>>(keys, f, sdim, n, inv, fs, dsrt);

  // 4) build packed GF(2) boundary matrix
  size_t nwords = (size_t)n * W32;
  k_zero<<<(unsigned)((nwords + 255) / 256), 256, 0, stream>>>(B, nwords);
  k_scatter<<<(m + 255) / 256, 256, 0, stream>>>(inv, bc, bf, m, W32, B);

  // 5) persistence reduction (sequential in j, wave-parallel row ops,
  //    async-LDS double buffering of the next row)
  k_reduce<<<1, 256, 0, stream>>>(B, n, W32, piv);

  // 6) losses via fp32 WMMA reduction
  k_loss<<<1, 64, 0, stream>>>(fs, dsrt, piv, n, (float*)d_out);
}